// FuzzyNet_66657892434321
// MI455X (gfx1250) — compile-verified
//
#include <hip/hip_runtime.h>
#include <hip/hip_bf16.h>

// FuzzyNet forward, CDNA5 (gfx1250), wave32.
//
// op[d,n,c]  = 10*sigmoid( sum_h W2[d,n,c,h]*relu(W1[d,n,h]*x[n]+b1[d,n,h]) + b2[d,n,c] )
// weights[i] = (sigmoid(w_weight[i]+w_bias[i])+1)/2
//
// Memory-bound: W2 is 512 MiB fp32 streamed once -> ~23 us floor @ 23.3 TB/s.
// The GEMV is mapped onto V_WMMA_F32_16X16X4_F32 with the h-vector broadcast
// across all 16 B-columns (column waste is free; we're bandwidth-bound).

typedef __attribute__((ext_vector_type(2))) float v2f;
typedef __attribute__((ext_vector_type(8))) float v8f;

#define D_DMS 32
#define N_NET 256
#define HID   64
#define NCLS  256

__device__ __forceinline__ float sigmoid10(float v) {
    return 10.0f / (1.0f + __expf(-v));
}

__global__ __launch_bounds__(256) void fuzzy_fwd_kernel(
    const float* __restrict__ x,        // [N]
    const float* __restrict__ W1,       // [D,N,H]
    const float* __restrict__ b1,       // [D,N,H]
    const float* __restrict__ W2,       // [D,N,C,H]
    const float* __restrict__ b2,       // [D,N,C]
    const float* __restrict__ w_weight, // [6]
    const float* __restrict__ w_bias,   // [6]
    float* __restrict__ out)            // [D*N*C] op, then [6] weights
{
    __shared__ float hbuf[HID];

    const int dn  = blockIdx.x;          // 0 .. D*N-1, encodes (d,n)
    const int n   = dn & (N_NET - 1);
    const int tid = threadIdx.x;

    // ---- hidden layer: h = relu(W1*x + b1), 64 values into LDS ----
    if (tid < HID) {
        const size_t base = (size_t)dn * HID + tid;
        float hv = W1[base] * x[n] + b1[base];
        hbuf[tid] = hv > 0.0f ? hv : 0.0f;
    }

    // ---- tail output: 6 mixing weights (block 0 only) ----
    if (dn == 0 && tid < 6) {
        float s = 1.0f / (1.0f + __expf(-(w_weight[tid] + w_bias[tid])));
        out[(size_t)D_DMS * N_NET * NCLS + tid] = (s + 1.0f) * 0.5f;
    }
    __syncthreads();

    // ---- GEMV via WMMA f32 16x16x4 ----
    // A layout (32-bit A 16x4): lanes 0-15 = M, VGPR0 holds K={0|2}, VGPR1 K={1|3}
    // with the K pair selected by the lane half.  B (4x16) mirrors this with
    // M<->N swapped; we broadcast h along the N dimension so only the K-per-
    // lane-half mapping matters.
    const int lane = tid & 31;
    const int wave = tid >> 5;           // 0..7
    const int half = lane >> 4;          // 0 | 1
    const int m    = lane & 15;          // row within the 16-wide c-tile

    const int c0a = wave * 32;           // first c-tile of this wave
    const int c0b = c0a + 16;            // second c-tile

    const size_t w2base = (size_t)dn * NCLS * HID;
    const float* rowA = W2 + w2base + (size_t)(c0a + m) * HID + 2 * half;
    const float* rowB = W2 + w2base + (size_t)(c0b + m) * HID + 2 * half;

    v8f acc0 = {0.f, 0.f, 0.f, 0.f, 0.f, 0.f, 0.f, 0.f};
    v8f acc1 = {0.f, 0.f, 0.f, 0.f, 0.f, 0.f, 0.f, 0.f};

#pragma unroll
    for (int k0 = 0; k0 < HID; k0 += 4) {
        // B fragment: h[k] broadcast across all 16 columns
        v2f bfrag;
        bfrag.x = hbuf[k0 + 2 * half];
        bfrag.y = hbuf[k0 + 2 * half + 1];
        // A fragments: 8B per lane, contiguous walk along each 256B W2 row
        v2f a0 = *(const v2f*)(rowA + k0);
        v2f a1 = *(const v2f*)(rowB + k0);
        // 8-arg form: (neg_a, A, neg_b, B, c_mod, C, reuse_a, reuse_b)
        acc0 = __builtin_amdgcn_wmma_f32_16x16x4_f32(
            false, a0, false, bfrag, (short)0, acc0, false, false);
        acc1 = __builtin_amdgcn_wmma_f32_16x16x4_f32(
            false, a1, false, bfrag, (short)0, acc1, false, false);
    }

    // ---- epilogue ----
    // D layout (32-bit C/D 16x16): lane holds column n = lane&15; VGPR r holds
    // row M = r + 8*half.  All columns are identical (B was broadcast), so the
    // n==0 lanes (lane 0 and lane 16) carry the full result: 8 rows each.
    if (m == 0) {
        const size_t obase = (size_t)dn * NCLS;
        const int cb0 = c0a + 8 * half;
        const int cb1 = c0b + 8 * half;
#pragma unroll
        for (int r = 0; r < 8; ++r) {
            float v = acc0[r] + b2[obase + cb0 + r];
            out[obase + cb0 + r] = sigmoid10(v);
        }
#pragma unroll
        for (int r = 0; r < 8; ++r) {
            float v = acc1[r] + b2[obase + cb1 + r];
            out[obase + cb1 + r] = sigmoid10(v);
        }
    }
}

extern "C" void kernel_launch(void* const* d_in, const int* in_sizes, int n_in,
                              void* d_out, int out_size, void* d_ws, size_t ws_size,
                              hipStream_t stream) {
    (void)in_sizes; (void)n_in; (void)d_ws; (void)ws_size; (void)out_size;

    const float* x        = (const float*)d_in[0];
    const float* W1       = (const float*)d_in[1];
    const float* b1       = (const float*)d_in[2];
    const float* W2       = (const float*)d_in[3];
    const float* b2       = (const float*)d_in[4];
    const float* w_weight = (const float*)d_in[5];
    const float* w_bias   = (const float*)d_in[6];
    float* out            = (float*)d_out;

    dim3 grid(D_DMS * N_NET);   // one workgroup per (dm, network)
    dim3 block(256);            // 8 waves: 8 * (2 c-tiles of 16) = 256 classes
    fuzzy_fwd_kernel<<<grid, block, 0, stream>>>(x, W1, b1, W2, b2,
                                                 w_weight, w_bias, out);
}